// GNN_Encoder_60567628808242
// MI455X (gfx1250) — compile-verified
//
#include <hip/hip_runtime.h>
#include <hip/hip_bf16.h>

// GCN 2-layer encoder for MI455X (gfx1250, wave32).
//
// Cost model: GEMMs are 6.5 GFLOP total (negligible); the edge gather +
// atomic scatter (2 x ~410MB) dominates, and the 51MB node-feature table is
// L2-resident (192MB L2), so edge phase is L2-atomic bound, not HBM bound.
// GEMM therefore stays exact fp32 via V_WMMA_F32_16X16X4_F32.

typedef __attribute__((ext_vector_type(2))) float v2f;
typedef __attribute__((ext_vector_type(8))) float v8f;

#define D 128  // D_IN == D_HID == D_OUT == 128

// ---------------- degree / normalization ----------------

__global__ __launch_bounds__(256) void k_deg_init(float* __restrict__ deg, int N) {
  int i = blockIdx.x * 256 + threadIdx.x;
  if (i < N) deg[i] = 1.0f;  // self loop
}

__global__ __launch_bounds__(256) void k_deg_count(const int* __restrict__ dst,
                                                   float* __restrict__ deg, int E) {
  int i = blockIdx.x * 256 + threadIdx.x;
  if (i < E) atomicAdd(&deg[dst[i]], 1.0f);
}

__global__ __launch_bounds__(256) void k_rsqrt(float* __restrict__ deg, int N) {
  int i = blockIdx.x * 256 + threadIdx.x;
  if (i < N) deg[i] = rsqrtf(deg[i]);  // deg >= 1 always (self loop)
}

__global__ __launch_bounds__(256) void k_norm(const int* __restrict__ src,
                                              const int* __restrict__ dst,
                                              const float* __restrict__ dinv,
                                              float* __restrict__ norm, int E) {
  int i = blockIdx.x * 256 + threadIdx.x;
  if (i < E) norm[i] = dinv[src[i]] * dinv[dst[i]];
}

// ---------------- fp32 WMMA GEMM: H[N,128] = X[N,128] @ W[128,128] ----------------
// Block = 256 threads = 8 waves. Block owns a 16-row M band; wave w owns
// 16-col N tile w. W staged once in LDS (64KB of the 320KB/WGP pool).

__global__ __launch_bounds__(256) void k_gemm(const float* __restrict__ X,
                                              const float* __restrict__ W,
                                              float* __restrict__ H, int N) {
  __shared__ float Wl[D * D];
  const int tid = threadIdx.x;
  for (int i = tid; i < D * D / 4; i += 256)
    ((float4*)Wl)[i] = ((const float4*)W)[i];
  __syncthreads();

  const int m0   = blockIdx.x * 16;
  const int lane = tid & 31;
  const int wave = tid >> 5;
  const int row  = lane & 15;          // A row / B-C column-within-tile
  const int kofs = (lane >> 4) << 1;   // 0 or 2: which K pair this half-wave holds
  const int col  = (wave << 4) + row;  // global output column

  int arow = m0 + row;                 // clamp reads; stores are guarded below
  if (arow >= N) arow = N - 1;
  const float2* xr = (const float2*)(X + (size_t)arow * D);

  v8f acc = {};
#pragma unroll
  for (int kb = 0; kb < D; kb += 4) {
    float2 af = xr[(kb + kofs) >> 1];        // A: row 'arow', K = kb+kofs .. +1
    v2f a; a.x = af.x; a.y = af.y;
    v2f b;                                   // B: K x col, lanes0-15 low-K half
    b.x = Wl[(kb + kofs) * D + col];
    b.y = Wl[(kb + kofs + 1) * D + col];
    acc = __builtin_amdgcn_wmma_f32_16x16x4_f32(
        /*neg_a=*/false, a, /*neg_b=*/false, b,
        /*c_mod=*/(short)0, acc, /*reuse_a=*/false, /*reuse_b=*/false);
  }

  const int mhi = (lane >> 4) << 3;          // C/D: M = i + 8*(lane>=16)
#pragma unroll
  for (int i = 0; i < 8; ++i) {
    int r = m0 + mhi + i;
    if (r < N) H[(size_t)r * D + col] = acc[i];
  }
}

// ---------------- out = dinv^2 * h + bias (self loop + bias, full overwrite) --------

__global__ __launch_bounds__(256) void k_self_bias(const float* __restrict__ H,
                                                   const float* __restrict__ dinv,
                                                   const float* __restrict__ bias,
                                                   float* __restrict__ out, int N) {
  int idx = blockIdx.x * 256 + threadIdx.x;  // one float4 chunk per thread
  if (idx >= N * (D / 4)) return;
  int v = idx >> 5;        // D/4 == 32 chunks per node
  int c = idx & 31;
  float di = dinv[v];
  float w = di * di;
  float4 h = ((const float4*)(H + (size_t)v * D))[c];
  float4 b = ((const float4*)bias)[c];
  float4 o = {w * h.x + b.x, w * h.y + b.y, w * h.z + b.z, w * h.w + b.w};
  ((float4*)(out + (size_t)v * D))[c] = o;
}

// ---------------- edge scatter: out[dst] += norm * h[src] ----------------
// One wave per edge; each lane moves a float4 (32 lanes * 16B = 512B/edge,
// coalesced). Atomic f32 adds resolve in the (L2-resident) destination rows.

__global__ __launch_bounds__(256) void k_scatter(const float* __restrict__ H,
                                                 const float* __restrict__ norm,
                                                 const int* __restrict__ src,
                                                 const int* __restrict__ dst,
                                                 float* __restrict__ out, int E) {
  int e = blockIdx.x * 8 + (threadIdx.x >> 5);
  if (e >= E) return;
  int lane = threadIdx.x & 31;
  int s = src[e];
  int d = dst[e];
  float w = norm[e];
  float4 v = ((const float4*)(H + (size_t)s * D))[lane];
  float* o = out + (size_t)d * D + lane * 4;
  atomicAdd(o + 0, w * v.x);
  atomicAdd(o + 1, w * v.y);
  atomicAdd(o + 2, w * v.z);
  atomicAdd(o + 3, w * v.w);
}

__global__ __launch_bounds__(256) void k_relu(float* __restrict__ buf, int n4) {
  int i = blockIdx.x * 256 + threadIdx.x;
  if (i >= n4) return;
  float4 v = ((float4*)buf)[i];
  v.x = fmaxf(v.x, 0.0f); v.y = fmaxf(v.y, 0.0f);
  v.z = fmaxf(v.z, 0.0f); v.w = fmaxf(v.w, 0.0f);
  ((float4*)buf)[i] = v;
}

// ---------------- host-side orchestration ----------------

static inline size_t align_up(size_t x, size_t a) { return (x + a - 1) & ~(a - 1); }

extern "C" void kernel_launch(void* const* d_in, const int* in_sizes, int n_in,
                              void* d_out, int out_size, void* d_ws, size_t ws_size,
                              hipStream_t stream) {
  const float* x  = (const float*)d_in[0];
  const float* W1 = (const float*)d_in[1];
  const float* b1 = (const float*)d_in[2];
  const float* W2 = (const float*)d_in[3];
  const float* b2 = (const float*)d_in[4];
  const int*   ei = (const int*)d_in[5];   // edge_index [2, E] flattened

  const int N = in_sizes[0] / D;   // 100000
  const int E = in_sizes[5] / 2;   // 800000
  const int* src = ei;
  const int* dst = ei + E;

  float* ws = (float*)d_ws;
  size_t off = 0;
  float* deg  = ws + off; off += align_up((size_t)N, 64);        // deg -> dinv in place
  float* nrm  = ws + off; off += align_up((size_t)E, 64);
  float* buf1 = ws + off; off += (size_t)N * D;                  // GEMM output
  float* buf2 = ws + off;                                        // layer-1 aggregate
  float* out  = (float*)d_out;

  const int gN   = (N + 255) / 256;
  const int gE   = (E + 255) / 256;
  const int gMM  = (N + 15) / 16;
  const int gNF  = (N * (D / 4) + 255) / 256;
  const int gSc  = (E + 7) / 8;

  // normalization
  k_deg_init <<<gN,  256, 0, stream>>>(deg, N);
  k_deg_count<<<gE,  256, 0, stream>>>(dst, deg, E);
  k_rsqrt    <<<gN,  256, 0, stream>>>(deg, N);                  // deg now holds dinv
  k_norm     <<<gE,  256, 0, stream>>>(src, dst, deg, nrm, E);

  // layer 1
  k_gemm     <<<gMM, 256, 0, stream>>>(x, W1, buf1, N);
  k_self_bias<<<gNF, 256, 0, stream>>>(buf1, deg, b1, buf2, N);
  k_scatter  <<<gSc, 256, 0, stream>>>(buf1, nrm, src, dst, buf2, E);
  k_relu     <<<gNF, 256, 0, stream>>>(buf2, N * (D / 4));

  // layer 2 (writes d_out; k_self_bias fully initializes it before atomics)
  k_gemm     <<<gMM, 256, 0, stream>>>(buf2, W2, buf1, N);
  k_self_bias<<<gNF, 256, 0, stream>>>(buf1, deg, b2, out, N);
  k_scatter  <<<gSc, 256, 0, stream>>>(buf1, nrm, src, dst, out, E);
  k_relu     <<<gNF, 256, 0, stream>>>(out, N * (D / 4));
}